// SoftTopk_31868657337013
// MI455X (gfx1250) — compile-verified
//
#include <hip/hip_runtime.h>

// CDNA5 / gfx1250 Sinkhorn soft-top-k.
// One workgroup per batch row; whole 200-iteration solve in registers.
// Scaling-form Sinkhorn (exp of the reference's log-domain recurrence),
// with mu folded into the dual scalings h_j = eg_j / mu:
//   E_ij = exp(-C_ij/eps)               (precomputed once per element)
//   u_i  = 1 / (h0*E0_i + h1*E1_i)
//   h_j  = (nu_j/mu) / sum_i u_i*E_ij   (nu0/mu = k, nu1/mu = n-k)
//   Gamma_ij = u_i * (mu*h_j) * E_ij
// Per-wave dual reduction (S0,S1) done with one V_WMMA_F32_16X16X4_F32.

typedef __attribute__((ext_vector_type(2))) float v2f;
typedef __attribute__((ext_vector_type(8))) float v8f;

#define EPS_F     0.01f
#define MAX_ITER  200
#define NCOL      8192
#define NTH       256
#define NWAVE     (NTH / 32)       // 8 waves
#define EPT       (NCOL / NTH)     // 32 elements per thread

__global__ __launch_bounds__(NTH, 1)
void sinkhorn_soft_topk_kernel(const float* __restrict__ scores,
                               const int*   __restrict__ kptr,
                               float*       __restrict__ out) {
    // red[parity][column j][wave]
    __shared__ __align__(16) float red[2][2][NWAVE];
    __shared__ float maxbuf[NWAVE];

    const int row  = blockIdx.x;
    const int tid  = threadIdx.x;
    const int lane = tid & 31;
    const int wave = tid >> 5;

    // ---- load 32 scores (8 x float4, coalesced) -------------------------
    const float4* src4 =
        (const float4*)(scores + (size_t)row * NCOL + (size_t)tid * EPT);
    float s[EPT];
#pragma unroll
    for (int i = 0; i < EPT / 4; ++i) {
        float4 v = src4[i];
        s[4 * i + 0] = v.x; s[4 * i + 1] = v.y;
        s[4 * i + 2] = v.z; s[4 * i + 3] = v.w;
    }

    // ---- row max of C_raw = max(s^2, (s-1)^2) ---------------------------
    float m = 0.0f;
#pragma unroll
    for (int i = 0; i < EPT; ++i) {
        float c0 = s[i] * s[i];
        float t  = s[i] - 1.0f;
        m = fmaxf(m, fmaxf(c0, t * t));
    }
#pragma unroll
    for (int off = 16; off > 0; off >>= 1)
        m = fmaxf(m, __shfl_xor(m, off, 32));
    if (lane == 0) maxbuf[wave] = m;
    __syncthreads();
    float M = maxbuf[0];
#pragma unroll
    for (int w = 1; w < NWAVE; ++w) M = fmaxf(M, maxbuf[w]);

    const float inv_me = 1.0f / (M * EPS_F);   // 1/(max * eps)

    // ---- Gibbs kernel entries E = exp(-C/eps), kept in registers --------
    float E0[EPT], E1[EPT], U[EPT];
#pragma unroll
    for (int i = 0; i < EPT; ++i) {
        float t = s[i] - 1.0f;
        E0[i] = __expf(-s[i] * s[i] * inv_me);
        E1[i] = __expf(-t * t * inv_me);
    }

    const int   kk  = *kptr;
    const float mu  = 1.0f / (float)NCOL;
    const float c0  = (float)kk;            // nu0/mu = k
    const float c1  = (float)(NCOL - kk);   // nu1/mu = n-k

    // WMMA selector B (4x16): rows K=0,2 feed columns n<8 (S0 path),
    // rows K=1,3 feed columns n>=8 (S1 path).
    v2f b;
    b.x = ((lane & 15) < 8) ? 1.0f : 0.0f;
    b.y = ((lane & 15) < 8) ? 0.0f : 1.0f;

    // store predicate: lanes {0,8} <=> (lane & 23) == 0
    const bool storer = (lane & 23) == 0;
    const int  jslot  = lane >> 3;          // 0 for lane 0, 1 for lane 8

    float h0 = (float)NCOL, h1 = (float)NCOL;   // eg_j/mu, g init = 0

#pragma unroll 2
    for (int it = 0; it < MAX_ITER; ++it) {
        // phase 1: denominators (pure FMA stream)
        float den[EPT];
#pragma unroll
        for (int i = 0; i < EPT; ++i)
            den[i] = fmaf(h0, E0[i], h1 * E1[i]);
        // phase 2: 32 independent v_rcp_f32 (trans pipe, latency hidden)
#pragma unroll
        for (int i = 0; i < EPT; ++i)
            U[i] = __builtin_amdgcn_rcpf(fmaxf(den[i], 1e-38f));
        // phase 3: accumulate S_j = sum u_i * E_ij
        float s0 = 0.0f, s1 = 0.0f;
#pragma unroll
        for (int i = 0; i < EPT; ++i) {
            s0 = fmaf(U[i], E0[i], s0);
            s1 = fmaf(U[i], E1[i], s1);
        }

        // --- dual cross-lane reduction on the matrix pipe ----------------
        // A(16x4): a.x/a.y per lane -> A[M][K0..K0+1], K0 = lane<16 ? 0 : 2
        // D[m][n<8]  = s0_m + s0_{m+16};  D[m][n>=8] = s1_m + s1_{m+16}
        v2f a; a.x = s0; a.y = s1;
        v8f c = {};
        v8f d = __builtin_amdgcn_wmma_f32_16x16x4_f32(
            /*neg_a=*/false, a, /*neg_b=*/false, b,
            /*c_mod=*/(short)0, c, /*reuse_a=*/false, /*reuse_b=*/false);
        float t = ((d[0] + d[1]) + (d[2] + d[3])) +
                  ((d[4] + d[5]) + (d[6] + d[7]));
        t += __shfl_xor(t, 16, 32);
        // lanes 0..7: wave S0 total; lanes 8..15: wave S1 total

        const int par = it & 1;      // double-buffered -> 1 barrier/iter
        if (storer) red[par][jslot][wave] = t;
        __syncthreads();

        // element-wise float4 accumulation -> clean v_pk_add_f32 codegen
        const float4* r = (const float4*)&red[par][0][0];
        float4 a0 = r[0];
        float4 a1 = r[NWAVE / 4];
#pragma unroll
        for (int q = 1; q < NWAVE / 4; ++q) {
            float4 v0 = r[q];
            float4 v1 = r[q + NWAVE / 4];
            a0.x += v0.x; a0.y += v0.y; a0.z += v0.z; a0.w += v0.w;
            a1.x += v1.x; a1.y += v1.y; a1.z += v1.z; a1.w += v1.w;
        }
        float S0 = (a0.x + a0.y) + (a0.z + a0.w);
        float S1 = (a1.x + a1.y) + (a1.z + a1.w);
        h0 = c0 * __builtin_amdgcn_rcpf(fmaxf(S0, 1e-38f));
        h1 = c1 * __builtin_amdgcn_rcpf(fmaxf(S1, 1e-38f));
    }

    // ---- Gamma = u * (mu*h_j) * E, interleaved [n][2] -------------------
    const float geg0 = mu * h0;
    const float geg1 = mu * h1;
    float2* out2 = (float2*)(out + (size_t)row * NCOL * 2) + (size_t)tid * EPT;
#pragma unroll
    for (int i = 0; i < EPT; ++i) {
        float2 g;
        g.x = U[i] * geg0 * E0[i];
        g.y = U[i] * geg1 * E1[i];
        out2[i] = g;
    }
}

extern "C" void kernel_launch(void* const* d_in, const int* in_sizes, int n_in,
                              void* d_out, int out_size, void* d_ws, size_t ws_size,
                              hipStream_t stream) {
    (void)n_in; (void)out_size; (void)d_ws; (void)ws_size;
    const float* scores = (const float*)d_in[0];
    const int*   kptr   = (const int*)d_in[1];
    float*       out    = (float*)d_out;
    const int batch = in_sizes[0] / NCOL;   // 512 rows of 8192
    sinkhorn_soft_topk_kernel<<<batch, NTH, 0, stream>>>(scores, kptr, out);
}